// OT_Attn_assem_4475355922967
// MI455X (gfx1250) — compile-verified
//
#include <hip/hip_runtime.h>
#include <math.h>

#define Nx 8192
#define My 4096
#define Dd 512
#define OT_REG 0.1f
#define OT_FI  0.8333333433f     /* tau/(tau+reg) = 0.5/0.6 */
#define A_MARG (1.0f / 8192.0f)
#define B_MARG (1.0f / 4096.0f)
#define NCHUNKS 32               /* j-chunks for u-update: 4096/128 */
#define UCHUNK  128
#define NITERS  100
#define FINBLOCKS 32768          /* MN / 1024 */

typedef __attribute__((ext_vector_type(2))) float v2f;
typedef __attribute__((ext_vector_type(8))) float v8f;

__device__ __forceinline__ float wave_sum(float v) {
  for (int m = 16; m >= 1; m >>= 1) v += __shfl_xor(v, m, 32);
  return v;
}
__device__ __forceinline__ float wave_max(float v) {
  for (int m = 16; m >= 1; m >>= 1) v = fmaxf(v, __shfl_xor(v, m, 32));
  return v;
}
__device__ __forceinline__ float wave_min(float v) {
  for (int m = 16; m >= 1; m >>= 1) v = fminf(v, __shfl_xor(v, m, 32));
  return v;
}

// ---------------- init u, v, cmax ----------------
__global__ void init_kernel(float* __restrict__ u, float* __restrict__ v,
                            unsigned* __restrict__ cmaxbits) {
  int t = blockIdx.x * 256 + threadIdx.x;
  if (t < Nx) u[t] = A_MARG;
  if (t < My) v[t] = B_MARG;
  if (t == 0) *cmaxbits = 0u;
}

// ---------------- row normalize: dst = src - rowmin, sq = sum(dst^2) ----------------
__global__ __launch_bounds__(256) void normalize_kernel(const float* __restrict__ src,
                                                        float* __restrict__ dst,
                                                        float* __restrict__ sq) {
  int lane = threadIdx.x & 31;
  int wave = threadIdx.x >> 5;
  int row  = blockIdx.x * 8 + wave;
  const float* s = src + (size_t)row * Dd;
  float4 r[4];
  float mn = 1e30f;
  for (int t = 0; t < 4; ++t) {
    r[t] = *(const float4*)(s + t * 128 + lane * 4);
    mn = fminf(mn, fminf(fminf(r[t].x, r[t].y), fminf(r[t].z, r[t].w)));
  }
  mn = wave_min(mn);
  float ss = 0.f;
  float* d = dst + (size_t)row * Dd;
  for (int t = 0; t < 4; ++t) {
    float4 o;
    o.x = r[t].x - mn; o.y = r[t].y - mn; o.z = r[t].z - mn; o.w = r[t].w - mn;
    ss += o.x * o.x + o.y * o.y + o.z * o.z + o.w * o.w;
    *(float4*)(d + t * 128 + lane * 4) = o;
  }
  ss = wave_sum(ss);
  if (lane == 0) sq[row] = ss;
}

// ---------------- costT[j,i] = max(x2_i + y2_j - 2*x_i.y_j, 0) via f32 WMMA ----------------
#define BI 128
#define BJ 64
#define KCH 32
#define LST 34  /* LDS row stride in floats: even (8B-aligned v2f) + bank spread */

__global__ __launch_bounds__(256) void gemm_cost_kernel(
    const float* __restrict__ xn, const float* __restrict__ yn,
    const float* __restrict__ x2, const float* __restrict__ y2,
    float* __restrict__ costT, unsigned* __restrict__ cmaxbits) {
  __shared__ __align__(16) float xs[BI * LST];
  __shared__ __align__(16) float ys[BJ * LST];
  __shared__ float wmx[8];

  const int tid = threadIdx.x;
  const int lane = tid & 31, w = tid >> 5;
  const int lm = lane & 15, lh = lane >> 4;
  const int wj = w & 1, wi = w >> 1;
  const int j0 = blockIdx.x * BJ;
  const int i0 = blockIdx.y * BI;

  v8f acc[2][2] = {};

  for (int k0 = 0; k0 < Dd; k0 += KCH) {
    // stage x tile: 128 rows x 32 cols ; y tile: 64 rows x 32 cols
    {
      int r = tid >> 3;            // 0..31
      int c = (tid & 7) * 4;       // 0..28
      for (int p = 0; p < 4; ++p) {
        int row = p * 32 + r;
        float4 val = *(const float4*)(xn + (size_t)(i0 + row) * Dd + k0 + c);
        float* q = &xs[row * LST + c];
        q[0] = val.x; q[1] = val.y; q[2] = val.z; q[3] = val.w;
      }
      for (int p = 0; p < 2; ++p) {
        int row = p * 32 + r;
        float4 val = *(const float4*)(yn + (size_t)(j0 + row) * Dd + k0 + c);
        float* q = &ys[row * LST + c];
        q[0] = val.x; q[1] = val.y; q[2] = val.z; q[3] = val.w;
      }
    }
    __syncthreads();
    for (int kk = 0; kk < KCH; kk += 4) {
      int col = kk + 2 * lh;  // even -> 8-byte aligned v2f in LDS
      v2f a0 = *(const v2f*)&ys[(wj * 32 + lm) * LST + col];
      v2f a1 = *(const v2f*)&ys[(wj * 32 + 16 + lm) * LST + col];
      v2f b0 = *(const v2f*)&xs[(wi * 32 + lm) * LST + col];
      v2f b1 = *(const v2f*)&xs[(wi * 32 + 16 + lm) * LST + col];
      acc[0][0] = __builtin_amdgcn_wmma_f32_16x16x4_f32(false, a0, false, b0, (short)0, acc[0][0], false, false);
      acc[0][1] = __builtin_amdgcn_wmma_f32_16x16x4_f32(false, a0, false, b1, (short)0, acc[0][1], false, false);
      acc[1][0] = __builtin_amdgcn_wmma_f32_16x16x4_f32(false, a1, false, b0, (short)0, acc[1][0], false, false);
      acc[1][1] = __builtin_amdgcn_wmma_f32_16x16x4_f32(false, a1, false, b1, (short)0, acc[1][1], false, false);
    }
    __syncthreads();
  }

  float tmax = 0.f;
  for (int js = 0; js < 2; ++js)
    for (int is = 0; is < 2; ++is) {
      int ii = i0 + wi * 32 + is * 16 + lm;
      float xv = x2[ii];
      int jb = j0 + wj * 32 + js * 16 + 8 * lh;
      for (int r = 0; r < 8; ++r) {
        int j = jb + r;
        float c = xv + y2[j] - 2.0f * acc[js][is][r];
        c = fmaxf(c, 0.0f);
        costT[(size_t)j * Nx + ii] = c;
        tmax = fmaxf(tmax, c);
      }
    }
  tmax = wave_max(tmax);
  if (lane == 0) wmx[w] = tmax;
  __syncthreads();
  if (tid == 0) {
    float m = wmx[0];
    for (int q = 1; q < 8; ++q) m = fmaxf(m, wmx[q]);
    atomicMax(cmaxbits, __float_as_uint(m));  // cost >= 0: uint order == float order
  }
}

// ---------------- KT = exp(-costT / (cmax*reg)) ----------------
__global__ void exp_kernel(const float* __restrict__ costT, float* __restrict__ KT,
                           const unsigned* __restrict__ cmaxbits) {
  float cmax = __uint_as_float(*cmaxbits);
  float s = -1.0f / (OT_REG * cmax);
  size_t p = ((size_t)blockIdx.x * 256 + threadIdx.x) * 4;
  float4 c = *(const float4*)(costT + p);
  float4 k;
  k.x = __expf(c.x * s); k.y = __expf(c.y * s);
  k.z = __expf(c.z * s); k.w = __expf(c.w * s);
  *(float4*)(KT + p) = k;
}

// ---------------- u-update part 1: upart[chunk][i] = sum_{j in chunk} KT[j,i]*v[j] ----------------
__global__ __launch_bounds__(256) void u_part_kernel(const float* __restrict__ KT,
                                                     const float* __restrict__ v,
                                                     float* __restrict__ upart) {
  int i0 = blockIdx.x * 1024 + threadIdx.x * 4;
  int jc = blockIdx.y;
  const float* base = KT + (size_t)jc * UCHUNK * Nx + i0;
  const float* vb = v + jc * UCHUNK;
  float4 acc = {0.f, 0.f, 0.f, 0.f};
  for (int j = 0; j < UCHUNK; ++j) {
    float vv = vb[j];  // uniform -> scalar load
    float4 k = *(const float4*)(base + (size_t)j * Nx);
    acc.x += k.x * vv; acc.y += k.y * vv; acc.z += k.z * vv; acc.w += k.w * vv;
  }
  *(float4*)(upart + (size_t)jc * Nx + i0) = acc;
}

// ---------------- u-update part 2: u_i = (a / sum_c upart)^fi ----------------
__global__ void u_final_kernel(const float* __restrict__ upart, float* __restrict__ u) {
  int i = blockIdx.x * 256 + threadIdx.x;
  float s = 0.f;
  for (int c = 0; c < NCHUNKS; ++c) s += upart[(size_t)c * Nx + i];
  u[i] = __expf(OT_FI * (__logf(A_MARG) - __logf(s)));
}

// ---------------- v-update: v_j = (b / (KT row_j . u))^fi ----------------
__global__ __launch_bounds__(256) void v_update_kernel(const float* __restrict__ KT,
                                                       const float* __restrict__ u,
                                                       float* __restrict__ v) {
  int lane = threadIdx.x & 31, w = threadIdx.x >> 5;
  int j = blockIdx.x * 8 + w;
  const float* row = KT + (size_t)j * Nx;
  float s = 0.f;
  for (int t = 0; t < Nx / 128; ++t) {
    int i = t * 128 + lane * 4;
    float4 k = *(const float4*)(row + i);
    float4 uu = *(const float4*)(u + i);
    s += k.x * uu.x + k.y * uu.y + k.z * uu.z + k.w * uu.w;
  }
  s = wave_sum(s);
  if (lane == 0) v[j] = __expf(OT_FI * (__logf(B_MARG) - __logf(s)));
}

// ---------------- pi[j,i] = u_i*KT[j,i]*v_j (in place over costT), dist partials ----------------
__global__ __launch_bounds__(256) void flow_kernel(float* __restrict__ out,
                                                   const float* __restrict__ KT,
                                                   const float* __restrict__ u,
                                                   const float* __restrict__ v,
                                                   float* __restrict__ distpart) {
  __shared__ float sred[8];
  size_t p4 = (size_t)blockIdx.x * 256 + threadIdx.x;
  size_t p = p4 * 4;
  int j = (int)(p >> 13);        // N = 8192 = 2^13
  int i = (int)(p & (Nx - 1));
  float vj = v[j];
  float4 k = *(const float4*)(KT + p);
  float4 uu = *(const float4*)(u + i);
  float4 c = *(const float4*)(out + p);
  float4 f;
  f.x = uu.x * k.x * vj; f.y = uu.y * k.y * vj;
  f.z = uu.z * k.z * vj; f.w = uu.w * k.w * vj;
  *(float4*)(out + p) = f;
  float part = c.x * f.x + c.y * f.y + c.z * f.z + c.w * f.w;
  part = wave_sum(part);
  if ((threadIdx.x & 31) == 0) sred[threadIdx.x >> 5] = part;
  __syncthreads();
  if (threadIdx.x == 0) {
    float t = sred[0];
    for (int q = 1; q < 8; ++q) t += sred[q];
    distpart[blockIdx.x] = t;
  }
}

__global__ void dist_reduce_kernel(const float* __restrict__ distpart,
                                   float* __restrict__ dist) {
  __shared__ float sred[8];
  float s = 0.f;
  for (int t = threadIdx.x; t < FINBLOCKS; t += 256) s += distpart[t];
  s = wave_sum(s);
  if ((threadIdx.x & 31) == 0) sred[threadIdx.x >> 5] = s;
  __syncthreads();
  if (threadIdx.x == 0) {
    float t = 0.f;
    for (int q = 0; q < 8; ++q) t += sred[q];
    *dist = t;
  }
}

extern "C" void kernel_launch(void* const* d_in, const int* in_sizes, int n_in,
                              void* d_out, int out_size, void* d_ws, size_t ws_size,
                              hipStream_t stream) {
  (void)in_sizes; (void)n_in; (void)out_size; (void)ws_size;
  const float* x = (const float*)d_in[0];
  const float* y = (const float*)d_in[1];
  float* out = (float*)d_out;
  float* W = (float*)d_ws;

  const size_t MN = (size_t)My * Nx;
  float* KT = W;                             // 134.2 MB, kept L2-resident across iters
  float* xn = KT + MN;
  float* yn = xn + (size_t)Nx * Dd;
  float* x2 = yn + (size_t)My * Dd;
  float* y2 = x2 + Nx;
  float* u  = y2 + My;
  float* v  = u + Nx;
  float* upart = v + My;
  float* distpart = upart + (size_t)NCHUNKS * Nx;
  unsigned* cmaxbits = (unsigned*)(distpart + FINBLOCKS);

  hipLaunchKernelGGL(init_kernel, dim3(Nx / 256), dim3(256), 0, stream, u, v, cmaxbits);
  hipLaunchKernelGGL(normalize_kernel, dim3(Nx / 8), dim3(256), 0, stream, x, xn, x2);
  hipLaunchKernelGGL(normalize_kernel, dim3(My / 8), dim3(256), 0, stream, y, yn, y2);
  // costT (M x N) written into d_out; max(cost) into cmaxbits
  hipLaunchKernelGGL(gemm_cost_kernel, dim3(My / BJ, Nx / BI), dim3(256), 0, stream,
                     xn, yn, x2, y2, out, cmaxbits);
  hipLaunchKernelGGL(exp_kernel, dim3((unsigned)(MN / 1024)), dim3(256), 0, stream,
                     out, KT, cmaxbits);
  for (int it = 0; it < NITERS; ++it) {
    hipLaunchKernelGGL(u_part_kernel, dim3(Nx / 1024, NCHUNKS), dim3(256), 0, stream,
                       KT, v, upart);
    hipLaunchKernelGGL(u_final_kernel, dim3(Nx / 256), dim3(256), 0, stream, upart, u);
    hipLaunchKernelGGL(v_update_kernel, dim3(My / 8), dim3(256), 0, stream, KT, u, v);
  }
  hipLaunchKernelGGL(flow_kernel, dim3(FINBLOCKS), dim3(256), 0, stream,
                     out, KT, u, v, distpart);
  hipLaunchKernelGGL(dist_reduce_kernel, dim3(1), dim3(256), 0, stream,
                     distpart, out + MN);
}